// ConvOffset3d_41704132444289
// MI455X (gfx1250) — compile-verified
//
#include <hip/hip_runtime.h>
#include <hip/hip_bf16.h>

// Problem constants (from reference)
#define N_    2
#define CIN_  32
#define D_    16
#define H_    32
#define W_    32
#define COUT_ 64
#define KD_   3
#define KH_   3
#define KW_   3
#define K_    27          // KD*KH*KW
#define CPG_  8
#define G_    4           // CIN / CPG
#define DO_   16
#define HO_   32
#define WO_   32
#define KC_   (CIN_ * K_) // 864 : reduction dim of the fused GEMM
#define TILE_N 16         // spatial positions per workgroup (one WMMA N tile)
#define P_IMG (DO_ * HO_ * WO_)   // 16384 positions per batch image
#define NP_   (N_ * P_IMG)        // 32768 total positions
#define CH_STRIDE (D_ * H_ * W_)  // 16384 floats between input channels

typedef __attribute__((ext_vector_type(2))) float v2f;
typedef __attribute__((ext_vector_type(8))) float v8f;

// One workgroup = 4 waves (wave32) = 128 threads.
// Phase 1: build col tile colB[kc=cin*27+k][p] (864x16 f32, 55.3 KB LDS) via
//          deformable trilinear gather (x lives in L2: only 4 MB).
// Phase 2: wave w computes the 16x16 D tile for couts [16w,16w+16) with
//          216 x V_WMMA_F32_16X16X4_F32, A streamed from weight (contiguous
//          rows, hot in L2/WGP$), B from LDS (bank-conflict-free layout).
__launch_bounds__(128, 2)
__global__ void deform_conv3d_wmma_kernel(const float* __restrict__ x,
                                          const float* __restrict__ offset,
                                          const float* __restrict__ weight,
                                          float* __restrict__ out) {
  __shared__ float colB[KC_ * TILE_N];  // 55296 bytes

  const int tile = blockIdx.x;          // 0 .. NP_/16 - 1
  const int p0   = tile * TILE_N;
  const int n    = p0 / P_IMG;
  const int pi   = p0 % P_IMG;          // in-image flat position of tile base
  const int dz   = pi / (HO_ * WO_);
  const int rem  = pi % (HO_ * WO_);
  const int hy   = rem / WO_;
  const int wx0  = rem % WO_;           // 0 or 16 (WO=32, tile=16)

  const int tid = threadIdx.x;

  // ---------------- Phase 1: deformable gather into LDS ----------------
  const float* __restrict__ xn   = x      + (size_t)n * CIN_ * CH_STRIDE;
  const float* __restrict__ offn = offset + (size_t)n * (G_ * K_ * 3) * P_IMG;

  for (int s = tid; s < G_ * K_ * TILE_N; s += 128) {   // 1728 samples / tile
    const int g  = s / (K_ * TILE_N);
    const int r  = s % (K_ * TILE_N);
    const int k  = r / TILE_N;
    const int p  = r % TILE_N;
    const int wx = wx0 + p;
    const int kd = k / (KH_ * KW_);
    const int kh = (k / KW_) % KH_;
    const int kw = k % KW_;

    const int sp = (dz * HO_ + hy) * WO_ + wx;
    const int ob = ((g * K_ + k) * 3) * P_IMG + sp;
    const float od = offn[ob];
    const float oh = offn[ob + P_IMG];
    const float ow = offn[ob + 2 * P_IMG];

    // stride 1, pad 1
    const float pd = (float)(dz - 1 + kd) + od;
    const float ph = (float)(hy - 1 + kh) + oh;
    const float pw = (float)(wx - 1 + kw) + ow;

    const float fd = floorf(pd), fh = floorf(ph), fw = floorf(pw);
    const int   d0 = (int)fd,    h0 = (int)fh,    w0 = (int)fw;
    const float ld = pd - fd,    lh = ph - fh,    lw = pw - fw;

    float acc[CPG_];
#pragma unroll
    for (int c = 0; c < CPG_; ++c) acc[c] = 0.0f;

    const float* __restrict__ xg = xn + (size_t)(g * CPG_) * CH_STRIDE;
#pragma unroll
    for (int di = 0; di < 2; ++di) {
#pragma unroll
      for (int hi = 0; hi < 2; ++hi) {
#pragma unroll
        for (int wi = 0; wi < 2; ++wi) {
          const int dd = d0 + di, hh = h0 + hi, ww = w0 + wi;
          const bool ok = (dd >= 0) & (dd < D_) & (hh >= 0) & (hh < H_) &
                          (ww >= 0) & (ww < W_);
          const float wgt = (di ? ld : 1.0f - ld) *
                            (hi ? lh : 1.0f - lh) *
                            (wi ? lw : 1.0f - lw);
          if (ok) {
            const float* __restrict__ xp = xg + ((size_t)dd * H_ + hh) * W_ + ww;
#pragma unroll
            for (int c = 0; c < CPG_; ++c)
              acc[c] = fmaf(wgt, xp[(size_t)c * CH_STRIDE], acc[c]);
          }
        }
      }
    }
    // colB[kc][p] with kc = (g*CPG + c)*K + k  == cin*27 + k
    // => weight A rows are contiguous: wr[o][kc] = weight[o*KC_ + kc]
#pragma unroll
    for (int c = 0; c < CPG_; ++c)
      colB[((g * CPG_ + c) * K_ + k) * TILE_N + p] = acc[c];
  }
  __syncthreads();

  // ---------------- Phase 2: WMMA GEMM (f32 16x16x4) ----------------
  const int wave  = tid >> 5;        // 0..3  -> cout tile
  const int lane  = tid & 31;
  const int o0    = wave * 16;
  const int mrow  = lane & 15;       // A: M row / B: N column / D: N column
  const int khalf = (lane >> 4) * 2; // K sub-offset: lanes 0-15 -> {0,1}, 16-31 -> {2,3}

  const float* __restrict__ wrow = weight + (size_t)(o0 + mrow) * KC_;

  v8f acc = {};
  for (int kc = 0; kc < KC_; kc += 4) {
    // A 16x4 f32: VGPR0 = K = khalf, VGPR1 = K = khalf+1 (contiguous -> b64 load)
    v2f a = *(const v2f*)(wrow + kc + khalf);
    // B 4x16 f32: same half-lane K split; rows 64B apart in LDS -> two ds_load_b32
    v2f b;
    b.x = colB[(kc + khalf + 0) * TILE_N + mrow];
    b.y = colB[(kc + khalf + 1) * TILE_N + mrow];
    acc = __builtin_amdgcn_wmma_f32_16x16x4_f32(
        /*neg_a=*/false, a, /*neg_b=*/false, b,
        /*c_mod=*/(short)0, acc, /*reuse_a=*/false, /*reuse_b=*/false);
  }

  // D 16x16 f32 layout: VGPR v holds M = v + 8*(lane>=16), N = lane&15
  float* __restrict__ op = out + (size_t)n * COUT_ * P_IMG + pi;
#pragma unroll
  for (int v = 0; v < 8; ++v) {
    const int m = v + 8 * (lane >> 4);
    op[(size_t)(o0 + m) * P_IMG + mrow] = acc[v];
  }
}

extern "C" void kernel_launch(void* const* d_in, const int* in_sizes, int n_in,
                              void* d_out, int out_size, void* d_ws, size_t ws_size,
                              hipStream_t stream) {
  const float* x      = (const float*)d_in[0];
  const float* offset = (const float*)d_in[1];
  const float* weight = (const float*)d_in[2];
  float* out          = (float*)d_out;
  (void)in_sizes; (void)n_in; (void)out_size; (void)d_ws; (void)ws_size;

  dim3 grid(NP_ / TILE_N);   // 2048 workgroups
  dim3 block(128);           // 4 wave32 waves
  deform_conv3d_wmma_kernel<<<grid, block, 0, stream>>>(x, offset, weight, out);
}